// IWT_UpSampling_Db4_58256936403048
// MI455X (gfx1250) — compile-verified
//
#include <hip/hip_runtime.h>
#include <hip/hip_bf16.h>

typedef __attribute__((ext_vector_type(2))) float v2f;
typedef __attribute__((ext_vector_type(8))) float v8f;

// db4 reconstruction taps, split by output parity.
// y[2m]   = sum_j CAE[j]*cA[m+j] + CDE[j]*cD[m+j]
// y[2m+1] = sum_j CAO[j]*cA[m+j] + CDO[j]*cD[m+j]
// (CAE[j]=REC_LO[6-2j], CAO[j]=REC_LO[7-2j], CDE/CDO from REC_HI)
__device__ __forceinline__ constexpr float kCAE(int j) {
  constexpr float t[4] = {0.032883011666982945f, -0.18703481171888114f,
                          0.6308807679295904f,    0.23037781330885523f};
  return t[j];
}
__device__ __forceinline__ constexpr float kCAO(int j) {
  constexpr float t[4] = {-0.010597401784997278f, 0.030841381835986965f,
                          -0.02798376941698385f,   0.7148465705525415f};
  return t[j];
}
__device__ __forceinline__ constexpr float kCDE(int j) {
  constexpr float t[4] = {0.7148465705525415f,   -0.02798376941698385f,
                          0.030841381835986965f, -0.010597401784997278f};
  return t[j];
}
__device__ __forceinline__ constexpr float kCDO(int j) {
  constexpr float t[4] = {-0.23037781330885523f, -0.6308807679295904f,
                          0.18703481171888114f,  -0.032883011666982945f};
  return t[j];
}

// Select one of 4 immediates by runtime index j; 0 outside [0,3].
__device__ __forceinline__ float sel4(int j, float x0, float x1, float x2, float x3) {
  float r = (j == 0) ? x0 : (j == 1) ? x1 : (j == 2) ? x2 : x3;
  return (j >= 0 && j <= 3) ? r : 0.0f;
}

#define TB 256

__global__ __launch_bounds__(TB) void idwt2_fused_kernel(const float* __restrict__ x,
                                                         float* __restrict__ out) {
  const int b   = blockIdx.y;          // batch 0..63
  const int bt  = blockIdx.x;          // tile 0..255
  const int bh  = bt >> 4;             // 0..15  (h tile)
  const int bw  = bt & 15;             // 0..15  (w tile)
  const int mh0 = bh * 16;             // first input h row of tile (pre-halo)
  const int mw0 = bw * 16;
  const int tid = threadIdx.x;

  __shared__ __align__(16) float sIn[19][19][16];  // input halo tile
  __shared__ float sLo[32][20][4];                 // stage-1 lo (col 19 = zero pad)
  __shared__ float sHi[32][20][4];                 // stage-1 hi
  __shared__ __align__(16) float sOut[32][32][4];  // staged output tile

  // ---------------- load 19x19x16 input halo (periodic), B128 ----------------
  for (int i = tid; i < 19 * 19 * 4; i += TB) {
    int qc = i & 3;            // which float4 of the 16 channels
    int s  = (i >> 2) % 19;    // w within tile
    int r  = i / 76;           // h within tile
    int hin = mh0 + r; hin -= (hin >= 256) ? 256 : 0;
    int win = mw0 + s; win -= (win >= 256) ? 256 : 0;
    const float4 v = *reinterpret_cast<const float4*>(
        x + ((((size_t)b * 256 + hin) * 256 + win) << 4) + (qc << 2));
    *reinterpret_cast<float4*>(&sIn[r][s][qc << 2]) = v;
  }
  __syncthreads();

  // ---------------- stage 1: H-axis IDWT (VALU FMA, immediates) --------------
  for (int i = tid; i < 16 * 19 * 4; i += TB) {
    int c = i & 3;
    int w = (i >> 2) % 19;
    int m = i / 76;            // 0..15
    float loE = 0.f, loO = 0.f, hiE = 0.f, hiO = 0.f;
#pragma unroll
    for (int j = 0; j < 4; ++j) {
      float a = sIn[m + j][w][c];          // cA
      float h = sIn[m + j][w][c + 4];      // cH
      float v = sIn[m + j][w][c + 8];      // cV
      float d = sIn[m + j][w][c + 12];     // cD
      loE = fmaf(kCAE(j), a, fmaf(kCDE(j), h, loE));
      loO = fmaf(kCAO(j), a, fmaf(kCDO(j), h, loO));
      hiE = fmaf(kCAE(j), v, fmaf(kCDE(j), d, hiE));
      hiO = fmaf(kCAO(j), v, fmaf(kCDO(j), d, hiO));
    }
    sLo[2 * m][w][c]     = loE;
    sLo[2 * m + 1][w][c] = loO;
    sHi[2 * m][w][c]     = hiE;
    sHi[2 * m + 1][w][c] = hiO;
  }
  // zero the K=12 pad column so padded WMMA lanes multiply 0*0
  for (int i = tid; i < 32 * 4; i += TB) {
    sLo[i >> 2][19][i & 3] = 0.f;
    sHi[i >> 2][19][i & 3] = 0.f;
  }
  __syncthreads();

  // ---------------- stage 2: W-axis IDWT on the matrix pipe ------------------
  // out(16h x 16w) = A_lo(16x12) * B_lo(12x16) + A_hi(16x12) * B_hi(12x16)
  // as 3 chained K=4 v_wmma_f32_16x16x4_f32 per operand.
  const int wave = tid >> 5;
  const int lane = tid & 31;
  const int n    = lane & 15;   // N column (and A-matrix M row)
  const int half = lane >> 4;   // selects K pair {2*half, 2*half+1}
  const int q    = n >> 1;      // input w position within block
  const int p    = n & 1;       // output parity

  // Per-lane constant banded B matrices (computed once, held in VGPRs).
  v2f BLo[3], BHi[3];
#pragma unroll
  for (int ck = 0; ck < 3; ++ck) {
    int k0 = 4 * ck + 2 * half;
    int j0 = k0 - q, j1 = k0 + 1 - q;
    float le0 = sel4(j0, kCAE(0), kCAE(1), kCAE(2), kCAE(3));
    float lo0 = sel4(j0, kCAO(0), kCAO(1), kCAO(2), kCAO(3));
    float le1 = sel4(j1, kCAE(0), kCAE(1), kCAE(2), kCAE(3));
    float lo1 = sel4(j1, kCAO(0), kCAO(1), kCAO(2), kCAO(3));
    float he0 = sel4(j0, kCDE(0), kCDE(1), kCDE(2), kCDE(3));
    float ho0 = sel4(j0, kCDO(0), kCDO(1), kCDO(2), kCDO(3));
    float he1 = sel4(j1, kCDE(0), kCDE(1), kCDE(2), kCDE(3));
    float ho1 = sel4(j1, kCDO(0), kCDO(1), kCDO(2), kCDO(3));
    BLo[ck].x = p ? lo0 : le0;  BLo[ck].y = p ? lo1 : le1;
    BHi[ck].x = p ? ho0 : he0;  BHi[ck].y = p ? ho1 : he1;
  }

  // 16 tiles (2 h-blocks x 2 w-blocks x 4 channels), exactly 2 per wave.
  // Compile-time 2-iteration unroll: no divergent loop, EXEC statically
  // all-ones for every WMMA, and the two 6-WMMA accumulation chains are
  // independent so the scheduler can interleave them and hide LDS latency.
#pragma unroll
  for (int it = 0; it < 2; ++it) {
    const int t     = wave + 8 * it;
    const int c     = t & 3;
    const int wb    = (t >> 2) & 1;
    const int hb    = t >> 3;
    const int h0    = hb * 16;
    const int wbase = wb * 8;

    // Issue all A-operand LDS loads up front, then run the WMMA chain.
    v2f aL[3], aH[3];
#pragma unroll
    for (int ck = 0; ck < 3; ++ck) {
      const int wi = wbase + 4 * ck + 2 * half;  // max 18; wi+1 <= 19 (zero pad)
      aL[ck].x = sLo[h0 + n][wi][c];   aL[ck].y = sLo[h0 + n][wi + 1][c];
      aH[ck].x = sHi[h0 + n][wi][c];   aH[ck].y = sHi[h0 + n][wi + 1][c];
    }
    v8f acc = {0.f, 0.f, 0.f, 0.f, 0.f, 0.f, 0.f, 0.f};
#pragma unroll
    for (int ck = 0; ck < 3; ++ck) {
      acc = __builtin_amdgcn_wmma_f32_16x16x4_f32(false, aL[ck], false, BLo[ck],
                                                  (short)0, acc, false, false);
      acc = __builtin_amdgcn_wmma_f32_16x16x4_f32(false, aH[ck], false, BHi[ck],
                                                  (short)0, acc, false, false);
    }
    // D layout: VGPR r -> M = r + 8*half, N = lane%16
#pragma unroll
    for (int r = 0; r < 8; ++r)
      sOut[h0 + r + 8 * half][wb * 16 + n][c] = acc[r];
  }
  __syncthreads();

  // ---------------- coalesced float4 stores (4 channels packed) --------------
  const size_t obase = ((size_t)b * 512 + 32 * bh) * 512 + 32 * bw;
#pragma unroll
  for (int k = 0; k < 4; ++k) {
    int i = tid + k * TB;           // uniform trip count: 1024 = 4*TB
    int h = i >> 5, w = i & 31;
    float4 v = *reinterpret_cast<const float4*>(&sOut[h][w][0]);
    *reinterpret_cast<float4*>(out + ((obase + (size_t)h * 512 + w) << 2)) = v;
  }
}

extern "C" void kernel_launch(void* const* d_in, const int* in_sizes, int n_in,
                              void* d_out, int out_size, void* d_ws, size_t ws_size,
                              hipStream_t stream) {
  const float* x = (const float*)d_in[0];
  float* out = (float*)d_out;
  (void)in_sizes; (void)n_in; (void)out_size; (void)d_ws; (void)ws_size;
  // 16x16 spatial tiles per batch; batch on grid.y
  idwt2_fused_kernel<<<dim3(256, 64, 1), dim3(TB, 1, 1), 0, stream>>>(x, out);
}